// GlobalItemConv_89197880803443
// MI455X (gfx1250) — compile-verified
//
#include <hip/hip_runtime.h>
#include <stdint.h>

#define D_FEAT 256
#define RP_BLK 256           // threads for row_ptr build
#define LBLK   64            // threads per row-block: 2 wave32, float4/thread
#define CHUNK  64            // edges staged into LDS per round

// ---------------------------------------------------------------------------
// CDNA5 async global->LDS staging (gfx1250 ASYNCcnt path, CDNA5 ISA)
// LDS destination is the byte offset held in a VGPR plus the instruction offset; tracked by ASYNCcnt.
// ---------------------------------------------------------------------------
__device__ __forceinline__ uint32_t lds_off_of(const void* p) {
    // generic shared pointer: low 32 bits = workgroup-relative LDS byte offset
    return (uint32_t)(uintptr_t)p;
}

__device__ __forceinline__ void async_load_b32_to_lds(uint32_t lds_byte_off,
                                                      const void* gaddr) {
#if defined(__gfx1250__)
    asm volatile("global_load_async_to_lds_b32 %0, %1, off"
                 :
                 : "v"(lds_byte_off), "v"(gaddr)
                 : "memory");
#else
    (void)lds_byte_off; (void)gaddr;
#endif
}

__device__ __forceinline__ void wait_asynccnt0() {
#if defined(__gfx1250__)
    asm volatile("s_wait_asynccnt 0" ::: "memory");
#endif
}

// ---------------------------------------------------------------------------
// Build CSR row_ptr from sorted COO row array: row_ptr[r] = lower_bound(row, r)
// ---------------------------------------------------------------------------
__global__ void build_row_ptr_kernel(const int* __restrict__ row, int n_edges,
                                     int n_nodes, int* __restrict__ row_ptr) {
    int r = blockIdx.x * blockDim.x + threadIdx.x;
    if (r > n_nodes) return;
    int lo = 0, hi = n_edges;
    while (lo < hi) {
        int mid = (lo + hi) >> 1;
        if (row[mid] < r) lo = mid + 1; else hi = mid;
    }
    row_ptr[r] = lo;
}

// ---------------------------------------------------------------------------
// One block (2 wave32, 64 threads, float4 per thread) per output row:
//   acc[d] = sum_e vals[e] * h_in[col[e], d]   (edge meta staged async->LDS,
//   feature rows gathered with global_load_b128)
//   h = acc / max(||acc||_2, eps)
//   mode 0: acc_out = x + h ; mode 1: acc_out += h ; mode 2: acc_out=(acc_out+h)*0.25
// ---------------------------------------------------------------------------
__global__ __launch_bounds__(LBLK) void layer_kernel(
    const float* __restrict__ h_in, const float* __restrict__ vals,
    const int* __restrict__ col, const int* __restrict__ row_ptr,
    float* __restrict__ h_out, float* __restrict__ acc_out,
    const float* __restrict__ x, int mode) {

    __shared__ int   s_col[CHUNK];
    __shared__ float s_val[CHUNK];
    __shared__ float s_red[LBLK / 32];

    const int r = blockIdx.x;
    const int t = threadIdx.x;            // owns features 4t .. 4t+3
    const int start = row_ptr[r];
    const int end   = row_ptr[r + 1];

    float4 acc = make_float4(0.f, 0.f, 0.f, 0.f);

    for (int base = start; base < end; base += CHUNK) {
        const int n = min(CHUNK, end - base);

        // Stage (col, val) for this chunk straight into LDS (ASYNCcnt path).
        if (t < n) {
            async_load_b32_to_lds(lds_off_of(&s_col[t]), (const void*)(col  + base + t));
            async_load_b32_to_lds(lds_off_of(&s_val[t]), (const void*)(vals + base + t));
        }
        wait_asynccnt0();
        __syncthreads();

        // Warm L2/L0 for the rows we are about to gather (global_prefetch_b8).
        if (t < n) {
            __builtin_prefetch(&h_in[(size_t)s_col[t] * D_FEAT], 0, 1);
        }

        // Broadcast (c, v) from LDS; gather 1KB feature rows as b128 loads.
        int i = 0;
        for (; i + 3 < n; i += 4) {
            const int   c0 = s_col[i + 0], c1 = s_col[i + 1];
            const int   c2 = s_col[i + 2], c3 = s_col[i + 3];
            const float v0 = s_val[i + 0], v1 = s_val[i + 1];
            const float v2 = s_val[i + 2], v3 = s_val[i + 3];
            const float4 a0 = *(const float4*)(h_in + (size_t)c0 * D_FEAT + 4 * t);
            const float4 a1 = *(const float4*)(h_in + (size_t)c1 * D_FEAT + 4 * t);
            const float4 a2 = *(const float4*)(h_in + (size_t)c2 * D_FEAT + 4 * t);
            const float4 a3 = *(const float4*)(h_in + (size_t)c3 * D_FEAT + 4 * t);
            acc.x = fmaf(v0, a0.x, acc.x); acc.y = fmaf(v0, a0.y, acc.y);
            acc.z = fmaf(v0, a0.z, acc.z); acc.w = fmaf(v0, a0.w, acc.w);
            acc.x = fmaf(v1, a1.x, acc.x); acc.y = fmaf(v1, a1.y, acc.y);
            acc.z = fmaf(v1, a1.z, acc.z); acc.w = fmaf(v1, a1.w, acc.w);
            acc.x = fmaf(v2, a2.x, acc.x); acc.y = fmaf(v2, a2.y, acc.y);
            acc.z = fmaf(v2, a2.z, acc.z); acc.w = fmaf(v2, a2.w, acc.w);
            acc.x = fmaf(v3, a3.x, acc.x); acc.y = fmaf(v3, a3.y, acc.y);
            acc.z = fmaf(v3, a3.z, acc.z); acc.w = fmaf(v3, a3.w, acc.w);
        }
        for (; i < n; ++i) {
            const int   c = s_col[i];
            const float v = s_val[i];
            const float4 a = *(const float4*)(h_in + (size_t)c * D_FEAT + 4 * t);
            acc.x = fmaf(v, a.x, acc.x); acc.y = fmaf(v, a.y, acc.y);
            acc.z = fmaf(v, a.z, acc.z); acc.w = fmaf(v, a.w, acc.w);
        }
        __syncthreads();  // before next chunk overwrites the staging buffers
    }

    // ---- block-wide sum of squares: float4 dot, wave32 shuffle, 2-wave LDS ----
    float ss = acc.x * acc.x + acc.y * acc.y + acc.z * acc.z + acc.w * acc.w;
    #pragma unroll
    for (int off = 16; off > 0; off >>= 1) ss += __shfl_xor(ss, off, 32);
    if ((t & 31) == 0) s_red[t >> 5] = ss;
    __syncthreads();
    const float tot  = s_red[0] + s_red[1];
    const float norm = sqrtf(tot);
    const float inv  = 1.0f / fmaxf(norm, 1e-12f);

    const float4 hv = make_float4(acc.x * inv, acc.y * inv, acc.z * inv, acc.w * inv);

    const size_t idx4 = (size_t)r * (D_FEAT / 4) + t;   // float4 index
    ((float4*)h_out)[idx4] = hv;

    if (mode == 0) {
        const float4 xv = ((const float4*)x)[idx4];
        ((float4*)acc_out)[idx4] =
            make_float4(xv.x + hv.x, xv.y + hv.y, xv.z + hv.z, xv.w + hv.w);
    } else if (mode == 1) {
        float4 a = ((const float4*)acc_out)[idx4];
        ((float4*)acc_out)[idx4] =
            make_float4(a.x + hv.x, a.y + hv.y, a.z + hv.z, a.w + hv.w);
    } else {
        float4 a = ((const float4*)acc_out)[idx4];
        ((float4*)acc_out)[idx4] =
            make_float4((a.x + hv.x) * 0.25f, (a.y + hv.y) * 0.25f,
                        (a.z + hv.z) * 0.25f, (a.w + hv.w) * 0.25f);
    }
}

// ---------------------------------------------------------------------------
// Host launcher. Inputs (setup_inputs order): x [N*D] f32, vals [E] f32,
// row [E] i32 (sorted), col [E] i32. Output: [N*D] f32.
// ---------------------------------------------------------------------------
extern "C" void kernel_launch(void* const* d_in, const int* in_sizes, int n_in,
                              void* d_out, int out_size, void* d_ws, size_t ws_size,
                              hipStream_t stream) {
    const float* x    = (const float*)d_in[0];
    const float* vals = (const float*)d_in[1];
    const int*   row  = (const int*)d_in[2];
    const int*   col  = (const int*)d_in[3];

    const int n_nodes = in_sizes[0] / D_FEAT;
    const int n_edges = in_sizes[1];

    char*  ws      = (char*)d_ws;
    int*   row_ptr = (int*)ws;
    size_t off     = (((size_t)(n_nodes + 1) * sizeof(int)) + 255) & ~(size_t)255;
    const size_t hbytes = (size_t)n_nodes * D_FEAT * sizeof(float);
    float* hA  = (float*)(ws + off);
    float* hB  = (float*)(ws + off + hbytes);
    float* out = (float*)d_out;

    // 1) CSR offsets (row is sorted, no atomics needed)
    const int rp_blocks = (n_nodes + 1 + RP_BLK - 1) / RP_BLK;
    build_row_ptr_kernel<<<rp_blocks, RP_BLK, 0, stream>>>(row, n_edges, n_nodes, row_ptr);

    // 2) Three SpMM + L2-normalize layers; accumulation folded into d_out:
    //    layer 1: out = x + h1 ; layer 2: out += h2 ; layer 3: out = (out + h3) / 4
    layer_kernel<<<n_nodes, LBLK, 0, stream>>>(x,  vals, col, row_ptr, hA, out, x, 0);
    layer_kernel<<<n_nodes, LBLK, 0, stream>>>(hA, vals, col, row_ptr, hB, out, x, 1);
    layer_kernel<<<n_nodes, LBLK, 0, stream>>>(hB, vals, col, row_ptr, hA, out, x, 2);
}